// ShapeMatchLoss_824633721416
// MI455X (gfx1250) — compile-verified
//
#include <hip/hip_runtime.h>
#include <math.h>

// ---------------------------------------------------------------------------
// ShapeMatchLoss on MI455X (gfx1250, wave32)
//
// Algebraic reduction:  d[b,i,j] = Sgg[j] + Spp[i] - 2*C[i,j], where all three
// derive from the 3x3 second-moment matrix M[b] = sum_n x x^T of points[b].
// Pass 1: memory-bound streaming reduction (96 MB @ 23.3 TB/s ~ 4.3 us).
// Pass 2: tiny finalize; K = G M G^T (G = [Rp;Rg], 6x3) done with two
//         V_WMMA_F32_16X16X4_F32 ops per batch.
// ---------------------------------------------------------------------------

#define BPB         32      // blocks per batch in pass 1
#define RED_THREADS 256     // 8 wave32s
#define FIN_THREADS 128     // 4 wave32s

typedef __attribute__((ext_vector_type(2))) float v2f;
typedef __attribute__((ext_vector_type(8))) float v8f;

#if defined(__AMDGCN__) && __has_builtin(__builtin_amdgcn_wmma_f32_16x16x4_f32)
#define USE_WMMA_F32X4 1
#else
#define USE_WMMA_F32X4 0
#endif

// ---------------------------------------------------------------------------
// Pass 1: per-batch second moments via b128 streaming loads.
// partials[block][6] = {xx, yy, zz, xy, xz, yz}
// ---------------------------------------------------------------------------
__global__ void __launch_bounds__(RED_THREADS)
moments_kernel(const float* __restrict__ points, float* __restrict__ partials, int N)
{
    const int tid   = threadIdx.x;
    const int batch = blockIdx.x / BPB;
    const int blk   = blockIdx.x % BPB;

    const size_t base = (size_t)batch * 3u * (size_t)N;
    const float4* __restrict__ px = (const float4*)(points + base);
    const float4* __restrict__ py = (const float4*)(points + base + (size_t)N);
    const float4* __restrict__ pz = (const float4*)(points + base + 2u * (size_t)N);

    const int n4 = N >> 2;
    float sxx = 0.f, syy = 0.f, szz = 0.f, sxy = 0.f, sxz = 0.f, syz = 0.f;

    for (int i = blk * RED_THREADS + tid; i < n4; i += BPB * RED_THREADS) {
        float4 x = px[i];
        float4 y = py[i];
        float4 z = pz[i];
        sxx += x.x * x.x + x.y * x.y + x.z * x.z + x.w * x.w;
        syy += y.x * y.x + y.y * y.y + y.z * y.z + y.w * y.w;
        szz += z.x * z.x + z.y * z.y + z.z * z.z + z.w * z.w;
        sxy += x.x * y.x + x.y * y.y + x.z * y.z + x.w * y.w;
        sxz += x.x * z.x + x.y * z.y + x.z * z.z + x.w * z.w;
        syz += y.x * z.x + y.y * z.y + y.z * z.z + y.w * z.w;
    }

    __shared__ float red[6][RED_THREADS];
    red[0][tid] = sxx; red[1][tid] = syy; red[2][tid] = szz;
    red[3][tid] = sxy; red[4][tid] = sxz; red[5][tid] = syz;
    __syncthreads();

    for (int off = RED_THREADS / 2; off > 0; off >>= 1) {
        if (tid < off) {
            #pragma unroll
            for (int k = 0; k < 6; ++k) red[k][tid] += red[k][tid + off];
        }
        __syncthreads();
    }
    if (tid < 6) partials[(size_t)blockIdx.x * 6 + tid] = red[tid][0];
}

// ---------------------------------------------------------------------------
// Pass 2 helpers
// ---------------------------------------------------------------------------
__device__ __forceinline__ void quat2rot(const float* __restrict__ q, float R[3][3])
{
    float x = q[0], y = q[1], z = q[2], w = q[3];
    float inv = 1.0f / sqrtf(x * x + y * y + z * z + w * w);
    x *= inv; y *= inv; z *= inv; w *= inv;
    float xx = x * x, yy = y * y, zz = z * z;
    float xy = x * y, xz = x * z, yz = y * z;
    float wx = w * x, wy = w * y, wz = w * z;
    R[0][0] = 1.f - 2.f * (yy + zz); R[0][1] = 2.f * (xy - wz); R[0][2] = 2.f * (xz + wy);
    R[1][0] = 2.f * (xy + wz); R[1][1] = 1.f - 2.f * (xx + zz); R[1][2] = 2.f * (yz - wx);
    R[2][0] = 2.f * (xz - wy); R[2][1] = 2.f * (yz + wx); R[2][2] = 1.f - 2.f * (xx + yy);
}

__device__ __forceinline__ float sel3(int r, float a, float b, float c)
{
    return r == 0 ? a : (r == 1 ? b : (r == 2 ? c : 0.0f));
}
__device__ __forceinline__ float sel6(int r, float a, float b, float c, float d, float e, float f)
{
    return r == 0 ? a : (r == 1 ? b : (r == 2 ? c : (r == 3 ? d : (r == 4 ? e : (r == 5 ? f : 0.0f)))));
}

// ---------------------------------------------------------------------------
// Pass 2: reduce partials -> M[b]; K = G M G^T via two WMMA f32 16x16x4 ops;
// chamfer scalar out. 4 waves, 8 batches each.
// ---------------------------------------------------------------------------
__global__ void __launch_bounds__(FIN_THREADS)
finalize_kernel(const float* __restrict__ predquat, const float* __restrict__ gtquat,
                const float* __restrict__ partials, float* __restrict__ out, int B)
{
    __shared__ float mom[32][6];
    __shared__ float sub[32][4][6];
    __shared__ float waveacc[FIN_THREADS / 32];

    const int tid = threadIdx.x;

    // --- stage A: deterministic tree-sum of pass-1 partials -> mom[b][6] ---
    {
        const int batch = tid >> 2;       // 0..31
        const int s     = tid & 3;
        float a0 = 0.f, a1 = 0.f, a2 = 0.f, a3 = 0.f, a4 = 0.f, a5 = 0.f;
        if (batch < B) {
            for (int blk = s; blk < BPB; blk += 4) {
                const float* p = partials + (size_t)(batch * BPB + blk) * 6;
                a0 += p[0]; a1 += p[1]; a2 += p[2];
                a3 += p[3]; a4 += p[4]; a5 += p[5];
            }
        }
        sub[batch][s][0] = a0; sub[batch][s][1] = a1; sub[batch][s][2] = a2;
        sub[batch][s][3] = a3; sub[batch][s][4] = a4; sub[batch][s][5] = a5;
    }
    __syncthreads();
    if (tid < B) {
        #pragma unroll
        for (int k = 0; k < 6; ++k)
            mom[tid][k] = sub[tid][0][k] + sub[tid][1][k] + sub[tid][2][k] + sub[tid][3][k];
    }
    __syncthreads();

    // --- stage B: per-batch 6x6 Gram matrix K = G M G^T ---
    const int wave = tid >> 5;
    const int lane = tid & 31;
    const int row  = lane & 15;      // matrix row (A) / column (B,D) index
    const int half = lane >> 4;      // lane half selects K index pair
    const int nw   = FIN_THREADS / 32;
    const int bpw  = (B + nw - 1) / nw;
    const int bend0 = (wave + 1) * bpw;
    const int bend  = bend0 < B ? bend0 : B;

    float acc = 0.f;

    for (int b = wave * bpw; b < bend; ++b) {
        float Rp[3][3], Rg[3][3];
        quat2rot(predquat + 4 * b, Rp);
        quat2rot(gtquat + 4 * b, Rg);

        float G[6][3];
        #pragma unroll
        for (int i = 0; i < 3; ++i)
            #pragma unroll
            for (int k = 0; k < 3; ++k) { G[i][k] = Rp[i][k]; G[3 + i][k] = Rg[i][k]; }

        const float M00 = mom[b][0], M11 = mom[b][1], M22 = mom[b][2];
        const float M01 = mom[b][3], M02 = mom[b][4], M12 = mom[b][5];

        float kpp0, kpp1, kpp2, kgg0, kgg1, kgg2;
        float c00, c01, c02, c10, c11, c12, c20, c21, c22;

#if USE_WMMA_F32X4
        // --- WMMA1: U(3x6) = M(3x3,pad) * G^T(3x6,pad) ------------------
        // A 16x4 f32 layout: VGPR0 = K{0|2}, VGPR1 = K{1|3}; row = lane%16.
        v2f a1v, b1v;
        a1v.x = half ? sel3(row, M02, M12, M22) : sel3(row, M00, M01, M02); // K=2 | K=0
        a1v.y = half ? 0.f                      : sel3(row, M01, M11, M12); // K=3 | K=1
        // B 4x16 f32 layout (mirror of A): VGPR0 = K{0|2}, VGPR1 = K{1|3}; col = lane%16.
        b1v.x = half ? sel6(row, G[0][2], G[1][2], G[2][2], G[3][2], G[4][2], G[5][2])
                     : sel6(row, G[0][0], G[1][0], G[2][0], G[3][0], G[4][0], G[5][0]);
        b1v.y = half ? 0.f
                     : sel6(row, G[0][1], G[1][1], G[2][1], G[3][1], G[4][1], G[5][1]);
        v8f cz = {};
        v8f U = __builtin_amdgcn_wmma_f32_16x16x4_f32(
            false, a1v, false, b1v, (short)0, cz, false, false);

        // --- WMMA2: K(6x6) = G(6x3,pad) * U(3x6,pad) --------------------
        // A2 = G: identical per-lane packing to B1 (both put G[row][k] at lane=row).
        v2f a2v = b1v;
        // B2 = U from D of WMMA1: D VGPR m @ lane n<16 holds U[m][n].
        float u2 = __shfl(U[2], row);                 // U[2][lane-16] for upper half
        v2f b2v;
        b2v.x = half ? u2  : U[0];
        b2v.y = half ? 0.f : U[1];
        v8f K = __builtin_amdgcn_wmma_f32_16x16x4_f32(
            false, a2v, false, b2v, (short)0, cz, false, false);

        // Extract K[m][n] (m,n<6): D VGPR m @ lane n.
        kpp0 = __shfl(K[0], 0); kpp1 = __shfl(K[1], 1); kpp2 = __shfl(K[2], 2);
        kgg0 = __shfl(K[3], 3); kgg1 = __shfl(K[4], 4); kgg2 = __shfl(K[5], 5);
        c00 = __shfl(K[0], 3); c01 = __shfl(K[0], 4); c02 = __shfl(K[0], 5);
        c10 = __shfl(K[1], 3); c11 = __shfl(K[1], 4); c12 = __shfl(K[1], 5);
        c20 = __shfl(K[2], 3); c21 = __shfl(K[2], 4); c22 = __shfl(K[2], 5);
#else
        // Scalar fallback: T = G*M, then needed entries of T*G^T.
        float T[6][3];
        #pragma unroll
        for (int r = 0; r < 6; ++r) {
            T[r][0] = G[r][0] * M00 + G[r][1] * M01 + G[r][2] * M02;
            T[r][1] = G[r][0] * M01 + G[r][1] * M11 + G[r][2] * M12;
            T[r][2] = G[r][0] * M02 + G[r][1] * M12 + G[r][2] * M22;
        }
        #define DOT3(u, v) (u[0] * v[0] + u[1] * v[1] + u[2] * v[2])
        kpp0 = DOT3(T[0], G[0]); kpp1 = DOT3(T[1], G[1]); kpp2 = DOT3(T[2], G[2]);
        kgg0 = DOT3(T[3], G[3]); kgg1 = DOT3(T[4], G[4]); kgg2 = DOT3(T[5], G[5]);
        c00 = DOT3(T[0], G[3]); c01 = DOT3(T[0], G[4]); c02 = DOT3(T[0], G[5]);
        c10 = DOT3(T[1], G[3]); c11 = DOT3(T[1], G[4]); c12 = DOT3(T[1], G[5]);
        c20 = DOT3(T[2], G[3]); c21 = DOT3(T[2], G[4]); c22 = DOT3(T[2], G[5]);
        #undef DOT3
#endif
        // d[i][j] = Sgg[j] + Spp[i] - 2 C[i][j]; row[j] = min_i; loss = mean_j.
        float d00 = kgg0 + kpp0 - 2.f * c00;
        float d10 = kgg0 + kpp1 - 2.f * c10;
        float d20 = kgg0 + kpp2 - 2.f * c20;
        float d01 = kgg1 + kpp0 - 2.f * c01;
        float d11 = kgg1 + kpp1 - 2.f * c11;
        float d21 = kgg1 + kpp2 - 2.f * c21;
        float d02 = kgg2 + kpp0 - 2.f * c02;
        float d12 = kgg2 + kpp1 - 2.f * c12;
        float d22 = kgg2 + kpp2 - 2.f * c22;
        float r0 = fminf(d00, fminf(d10, d20));
        float r1 = fminf(d01, fminf(d11, d21));
        float r2 = fminf(d02, fminf(d12, d22));
        acc += (r0 + r1 + r2) * (1.0f / 3.0f);
    }

    if (lane == 0) waveacc[wave] = acc;
    __syncthreads();
    if (tid == 0) {
        float t = 0.f;
        #pragma unroll
        for (int w = 0; w < FIN_THREADS / 32; ++w) t += waveacc[w];
        out[0] = t / (float)B;
    }
}

// ---------------------------------------------------------------------------
// Launch
// ---------------------------------------------------------------------------
extern "C" void kernel_launch(void* const* d_in, const int* in_sizes, int n_in,
                              void* d_out, int out_size, void* d_ws, size_t ws_size,
                              hipStream_t stream)
{
    const float* predquat = (const float*)d_in[0];   // [B,4]
    const float* gtquat   = (const float*)d_in[1];   // [B,4]
    const float* points   = (const float*)d_in[2];   // [B,3,N]

    const int B = in_sizes[0] / 4;                   // 32
    const int N = (int)((long long)in_sizes[2] / (3LL * B)); // 262144

    float* partials = (float*)d_ws;                  // B*BPB*6 floats = 24 KB

    moments_kernel<<<dim3(B * BPB), RED_THREADS, 0, stream>>>(points, partials, N);
    finalize_kernel<<<dim3(1), FIN_THREADS, 0, stream>>>(predquat, gtquat, partials,
                                                         (float*)d_out, B);
}